// ImageROICut_55405078119270
// MI455X (gfx1250) — compile-verified
//
#include <hip/hip_runtime.h>
#include <cstdint>

#define IMG 512
#define CUT 56
#define FC 32      // feature channels
#define SC 9       // mask channels
#define THREADS 256
#define ROWS_PER_BLK 8

#ifndef __has_builtin
#define __has_builtin(x) 0
#endif

typedef int   v4i __attribute__((ext_vector_type(4)));
typedef float v4f __attribute__((ext_vector_type(4)));
typedef __attribute__((address_space(1))) v4i* g_v4i_ptr;   // global
typedef __attribute__((address_space(3))) v4i* l_v4i_ptr;   // LDS

// ---- CDNA5 async global->LDS copy (ASYNCcnt path) -------------------------
__device__ __forceinline__ void async_copy16(const float* g, void* l) {
#if __has_builtin(__builtin_amdgcn_global_load_async_to_lds_b128)
  __builtin_amdgcn_global_load_async_to_lds_b128(
      (g_v4i_ptr)(uintptr_t)g,                 // src: v4i addrspace(1)*
      (l_v4i_ptr)(uint32_t)(uintptr_t)l,       // dst: v4i addrspace(3)* (low 32 bits = LDS offset)
      0, 0);                                   // cpol=0 (RT: keep gather set L2-resident)
#else
  unsigned lo = (unsigned)(uintptr_t)l;
  asm volatile("global_load_async_to_lds_b128 %0, %1, off"
               :: "v"(lo), "v"(g)
               : "memory");
#endif
}

template <int N>
__device__ __forceinline__ void wait_asynccnt() {
#if __has_builtin(__builtin_amdgcn_s_wait_asynccnt)
  __builtin_amdgcn_s_wait_asynccnt(N);
#else
  asm volatile("s_wait_asynccnt %0" :: "n"(N));
#endif
  asm volatile("" ::: "memory");
}

// ---------------------------------------------------------------------------
template <int LC>   // LC>0: compile-time n_levels; LC==0: runtime Lrt
__global__ __launch_bounds__(THREADS) void ImageROICut_kernel(
    const float* __restrict__ feat,   // [B,512,512,32]
    const float* __restrict__ gmask,  // [B,512,512,9]
    const float* __restrict__ rois,   // [B*R,4] (x1,y1,x2,y2) in pixels
    float* __restrict__ outF,         // [B*R,56,56,32*L]
    float* __restrict__ outM,         // [B*R,56,56,9]
    int R, int Lrt)
{
  const int L = LC ? LC : Lrt;

  __shared__ int   sx0[CUT], sx1[CUT], sy0[CUT], sy1[CUT];
  __shared__ float slx[CUT], sly[CUT];
  __shared__ v4f   stage[2][4][THREADS];   // double-buffered gather staging (32 KB)

  const int pair   = blockIdx.x;
  const int b      = pair / R;
  const int t      = threadIdx.x;
  const int oyBase = blockIdx.y * ROWS_PER_BLK;

  // ROI -> normalized coords (TF crop_and_resize: corners map to pixel centers)
  const float x1n = rois[pair*4+0] * (1.0f/IMG);
  const float y1n = rois[pair*4+1] * (1.0f/IMG);
  const float x2n = rois[pair*4+2] * (1.0f/IMG);
  const float y2n = rois[pair*4+3] * (1.0f/IMG);

  if (t < CUT) {
    const float frac = (float)t * (1.0f/(CUT-1));
    const float iy = (y1n + (y2n - y1n)*frac) * (float)(IMG-1);
    const float ix = (x1n + (x2n - x1n)*frac) * (float)(IMG-1);
    const float yf = floorf(iy), xf = floorf(ix);
    int y0 = (int)yf; y0 = y0 < 0 ? 0 : (y0 > IMG-1 ? IMG-1 : y0);
    int x0 = (int)xf; x0 = x0 < 0 ? 0 : (x0 > IMG-1 ? IMG-1 : x0);
    sy0[t] = y0; sy1[t] = (y0+1 > IMG-1) ? IMG-1 : y0+1;
    sx0[t] = x0; sx1[t] = (x0+1 > IMG-1) ? IMG-1 : x0+1;
    sly[t] = iy - yf; slx[t] = ix - xf;
  }
  __syncthreads();

  const float* fb = feat  + (size_t)b * ((size_t)IMG*IMG*FC);
  const float* mb = gmask + (size_t)b * ((size_t)IMG*IMG*SC);
  float* of = outF + (size_t)pair * ((size_t)CUT*CUT*FC*L);
  float* om = outM + (size_t)pair * ((size_t)CUT*CUT*SC);

  // ================= features: async-staged, double-buffered =================
  constexpr int NTASK = ROWS_PER_BLK*CUT*(FC/4);  // 3584 (exact multiple of 256)
  constexpr int ITERS = NTASK / THREADS;          // 14

  auto issue = [&](int it, int buf) {
    const int task = it*THREADS + t;
    const int c4  = task & (FC/4 - 1);
    const int pix = task >> 3;
    const int ox  = pix % CUT;
    const int oy  = oyBase + pix / CUT;
    const int x0 = sx0[ox], x1 = sx1[ox];
    const int y0 = sy0[oy], y1 = sy1[oy];
    const int cb = c4*4;
    async_copy16(fb + ((y0*IMG + x0)*FC + cb), &stage[buf][0][t]);
    async_copy16(fb + ((y0*IMG + x1)*FC + cb), &stage[buf][1][t]);
    async_copy16(fb + ((y1*IMG + x0)*FC + cb), &stage[buf][2][t]);
    async_copy16(fb + ((y1*IMG + x1)*FC + cb), &stage[buf][3][t]);
  };

  issue(0, 0);
  for (int it = 0; it < ITERS; ++it) {
    const int buf = it & 1;
    if (it + 1 < ITERS) {
      issue(it+1, (it+1)&1);   // prefetch next tile's taps
      wait_asynccnt<4>();      // current tile's 4 copies complete (in-order)
    } else {
      wait_asynccnt<0>();
    }
    const v4f v00 = stage[buf][0][t];
    const v4f v01 = stage[buf][1][t];
    const v4f v10 = stage[buf][2][t];
    const v4f v11 = stage[buf][3][t];

    const int task = it*THREADS + t;
    const int c4  = task & (FC/4 - 1);
    const int pix = task >> 3;
    const int ox  = pix % CUT;
    const int oy  = oyBase + pix / CUT;
    const float lx = slx[ox], ly = sly[oy];
    const float omx = 1.0f - lx, omy = 1.0f - ly;

    const v4f top = v00*omx + v01*lx;
    const v4f bot = v10*omx + v11*lx;
    const v4f res = top*omy + bot*ly;

    // replicate across n_levels copies; NT stores: output is never re-read,
    // keep the 192MB L2 for the gather working set instead.
    float* dst = of + (size_t)(oy*CUT + ox) * (FC*L) + c4*4;
    #pragma unroll
    for (int l = 0; l < (LC ? LC : Lrt); ++l)
      __builtin_nontemporal_store(res, reinterpret_cast<v4f*>(dst + l*FC));
  }

  // ================= mask: direct scalar gather =============================
  const int MTASK = ROWS_PER_BLK*CUT*SC;   // 4032
  for (int task = t; task < MTASK; task += THREADS) {
    const int sc  = task % SC;
    const int pix = task / SC;
    const int ox  = pix % CUT;
    const int oy  = oyBase + pix / CUT;
    const int x0 = sx0[ox], x1 = sx1[ox];
    const int y0 = sy0[oy], y1 = sy1[oy];
    const float lx = slx[ox], ly = sly[oy];
    const float m00 = mb[(y0*IMG + x0)*SC + sc];
    const float m01 = mb[(y0*IMG + x1)*SC + sc];
    const float m10 = mb[(y1*IMG + x0)*SC + sc];
    const float m11 = mb[(y1*IMG + x1)*SC + sc];
    const float tp = m00*(1.0f-lx) + m01*lx;
    const float bt = m10*(1.0f-lx) + m11*lx;
    __builtin_nontemporal_store(tp*(1.0f-ly) + bt*ly,
                                &om[(size_t)(oy*CUT + ox)*SC + sc]);
  }
}

// ---------------------------------------------------------------------------
extern "C" void kernel_launch(void* const* d_in, const int* in_sizes, int n_in,
                              void* d_out, int out_size, void* d_ws, size_t ws_size,
                              hipStream_t stream) {
  (void)n_in; (void)d_ws; (void)ws_size;
  const float* feat  = (const float*)d_in[0];
  const float* gmask = (const float*)d_in[1];
  const float* rois  = (const float*)d_in[2];

  const int npair  = in_sizes[2] / 4;                 // B*R = 512
  const int B      = in_sizes[0] / (IMG*IMG*FC);      // 8
  const int R      = npair / B;                       // 64
  const int perPix = out_size / (npair * CUT * CUT);  // 32*L + 9
  const int L      = (perPix - SC) / FC;              // n_levels (4), host-derived

  float* outF = (float*)d_out;
  float* outM = outF + (size_t)npair * CUT * CUT * FC * L;

  dim3 grid(npair, CUT / ROWS_PER_BLK);   // 512 x 7 blocks
  if (L == 4) {
    hipLaunchKernelGGL((ImageROICut_kernel<4>), grid, dim3(THREADS), 0, stream,
                       feat, gmask, rois, outF, outM, R, L);
  } else {
    hipLaunchKernelGGL((ImageROICut_kernel<0>), grid, dim3(THREADS), 0, stream,
                       feat, gmask, rois, outF, outM, R, L);
  }
}